// SpatiallySparseConv_11132555231814
// MI455X (gfx1250) — compile-verified
//
#include <hip/hip_runtime.h>

// ---------------------------------------------------------------------------
// Spatially sparse conv as scatter-GEMM on gfx1250.
//   out[out_map[k,p]] += features[in_map[k,p]] @ weight[k]   (+ bias)
// fp32 end-to-end using V_WMMA_F32_16X16X4_F32 (precision matches reference;
// the op is L2-gather/atomic bound, so low-precision WMMA buys nothing).
// ---------------------------------------------------------------------------

typedef __attribute__((ext_vector_type(2))) float v2f;
typedef __attribute__((ext_vector_type(4))) float v4f;
typedef __attribute__((ext_vector_type(8))) float v8f;

#define K_OFFSETS 27
#define C_IN 32
#define C_OUT 32
#define WAVES_PER_BLOCK 8
#define TILES_PER_WAVE 4
#define TILE_ROWS 16
#define TILES_PER_BLOCK (WAVES_PER_BLOCK * TILES_PER_WAVE)

// out[i] = bias[i % 32]  (d_out is poisoned by the harness, must init)
__global__ __launch_bounds__(256) void sparse_conv_bias_init(
    float* __restrict__ out, const float* __restrict__ bias, int total) {
  int i = blockIdx.x * blockDim.x + threadIdx.x;
  if (i < total) out[i] = bias[i & (C_OUT - 1)];
}

__global__ __launch_bounds__(256) void sparse_conv_gemm(
    const float* __restrict__ features,   // [N_IN, 32]
    const float* __restrict__ weight,     // [27, 32, 32]
    const int*   __restrict__ in_map,     // [27, P]
    const int*   __restrict__ out_map,    // [27, P]
    float*       __restrict__ out,        // [N_OUT, 32]
    int P, int tiles_per_k) {
  __shared__ float Wlds[C_IN * C_OUT];    // 4 KB: weight[k] staged per block

  const int k   = blockIdx.y;
  const int tid = threadIdx.x;

  // Cooperative stage of weight[k] (1024 f32) into LDS: one float4 per thread.
  {
    const v4f* wsrc = (const v4f*)(weight + (long)k * (C_IN * C_OUT));
    ((v4f*)Wlds)[tid] = wsrc[tid];
  }
  __syncthreads();

  const int lane = tid & 31;
  const int wave = tid >> 5;
  const int half = lane >> 4;   // 0: K rows {0,1} / out rows 0-7 ; 1: {2,3} / 8-15
  const int lrow = lane & 15;   // A: matrix row ; B/C/D: column within 16-tile

  // B fragments (4x16 f32 per k-step): v0 = W[4kk+2h][col], v1 = W[4kk+2h+1][col]
  // Loaded once per block; reused across all tiles this wave processes.
  v2f Bfrag[2][8];
#pragma unroll
  for (int n = 0; n < 2; ++n) {
#pragma unroll
    for (int kk = 0; kk < 8; ++kk) {
      const int r = 4 * kk + 2 * half;
      v2f b;
      b.x = Wlds[(r + 0) * C_OUT + n * 16 + lrow];
      b.y = Wlds[(r + 1) * C_OUT + n * 16 + lrow];
      Bfrag[n][kk] = b;
    }
  }

  const int map_base = k * P;

#pragma unroll
  for (int t = 0; t < TILES_PER_WAVE; ++t) {
    // Tile index is wave-uniform -> EXEC stays all-ones around the WMMAs.
    const int tile = blockIdx.x * TILES_PER_BLOCK + wave * TILES_PER_WAVE + t;
    if (tile >= tiles_per_k) continue;
    const int p0 = tile * TILE_ROWS;

    // Prefetch the next tile's map entries into cache.
    __builtin_prefetch(in_map + map_base + p0 + TILE_ROWS, 0, 0);

    // Gather: lane owns feature row lrow of the tile; K split by lane half.
    const int fin = in_map[map_base + p0 + lrow];
    const float* frow = features + (long)fin * C_IN + 2 * half;

    v8f acc0 = {0.f, 0.f, 0.f, 0.f, 0.f, 0.f, 0.f, 0.f};
    v8f acc1 = acc0;
#pragma unroll
    for (int kk = 0; kk < 8; ++kk) {
      const v2f a = *(const v2f*)(frow + 4 * kk);   // K = 4kk+2h, 4kk+2h+1
      acc0 = __builtin_amdgcn_wmma_f32_16x16x4_f32(
          false, a, false, Bfrag[0][kk], (short)0, acc0, false, false);
      acc1 = __builtin_amdgcn_wmma_f32_16x16x4_f32(
          false, a, false, Bfrag[1][kk], (short)0, acc1, false, false);
    }

    // Scatter-add. C/D layout: lane = column (lrow / lrow+16),
    // VGPR v = output row 8*half + v. Rows collide globally -> f32 atomics.
    const int* orows = out_map + map_base + p0 + 8 * half;
#pragma unroll
    for (int v = 0; v < 8; ++v) {
      const int oidx = orows[v];
      float* obase = out + (long)oidx * C_OUT + lrow;
      atomicAdd(obase,      acc0[v]);
      atomicAdd(obase + 16, acc1[v]);
    }
  }
}

extern "C" void kernel_launch(void* const* d_in, const int* in_sizes, int n_in,
                              void* d_out, int out_size, void* d_ws, size_t ws_size,
                              hipStream_t stream) {
  const float* features = (const float*)d_in[0];
  const float* weight   = (const float*)d_in[1];
  const float* bias     = (const float*)d_in[2];
  const int*   in_map   = (const int*)d_in[3];
  const int*   out_map  = (const int*)d_in[4];
  float*       out      = (float*)d_out;

  const int P = in_sizes[3] / K_OFFSETS;                 // 150000
  const int tiles_per_k = (P + TILE_ROWS - 1) / TILE_ROWS;

  // 1) out = bias (broadcast over rows)
  sparse_conv_bias_init<<<(out_size + 255) / 256, 256, 0, stream>>>(
      out, bias, out_size);

  // 2) scatter-GEMM: grid.y = kernel offset, grid.x = P-chunks of 512 rows
  dim3 grid((tiles_per_k + TILES_PER_BLOCK - 1) / TILES_PER_BLOCK, K_OFFSETS);
  sparse_conv_gemm<<<grid, 256, 0, stream>>>(
      features, weight, in_map, out_map, out, P, tiles_per_k);
}